// StructuredAttention_29618094473319
// MI455X (gfx1250) — compile-verified
//
#include <hip/hip_runtime.h>
#include <math.h>
#include <stdint.h>

// Problem constants
#define B_  32
#define L_  512
#define H_  768
#define T_  8
#define DK_ 96
#define NB_ 16   // Gauss-Jordan block size
#define PSTR 20  // panel LDS stride (floats): 80B rows -> 16B-aligned chunks, conflict-free

typedef float v2f __attribute__((ext_vector_type(2)));
typedef float v8f __attribute__((ext_vector_type(8)));

// ---------------------------------------------------------------------------
// Kernel 1: fused Q/K projection.  out = x @ W^T + bias, stored head-major:
//   out[((b*T + t)*L + l)*DK + d],  n = t*DK + d
// One wave computes a 16x16 tile of the (B*L) x H GEMM via V_WMMA_F32_16X16X4_F32.
// blockIdx.z selects Q (0) or K (1).
// ---------------------------------------------------------------------------
__global__ __launch_bounds__(32)
void proj_kernel(const float* __restrict__ x,
                 const float* __restrict__ Wq, const float* __restrict__ bq,
                 const float* __restrict__ Wk, const float* __restrict__ bk,
                 float* __restrict__ qout, float* __restrict__ kout)
{
    const float* W    = (blockIdx.z == 0) ? Wq : Wk;
    const float* bias = (blockIdx.z == 0) ? bq : bk;
    float*       out  = (blockIdx.z == 0) ? qout : kout;

    const int n0   = blockIdx.x * 16;        // output column tile (0..H-1)
    const int m0   = blockIdx.y * 16;        // output row tile over B*L
    const int lane = threadIdx.x;
    const int lm   = lane & 15;
    const int kh   = (lane >> 4) * 2;        // K sub-offset: 0 (lanes 0-15) / 2 (16-31)

    const float* arow = x + (size_t)(m0 + lm) * H_;   // A operand row (x)
    const float* brow = W + (size_t)(n0 + lm) * H_;   // B operand: B[k][n] = W[n][k]

    v8f acc = {};
    for (int k0 = 0; k0 < H_; k0 += 4) {
        v2f a, b;
        a.x = arow[k0 + kh];
        a.y = arow[k0 + kh + 1];
        b.x = brow[k0 + kh];
        b.y = brow[k0 + kh + 1];
        acc = __builtin_amdgcn_wmma_f32_16x16x4_f32(false, a, false, b,
                                                    (short)0, acc, false, false);
    }

    const int n  = n0 + lm;
    const float bv = bias[n];
    const int t  = n / DK_;
    const int d  = n - t * DK_;
    const int bb = m0 / L_;                  // 16-row tile never crosses batch (512 % 16 == 0)
    const int l0 = m0 % L_;
    float* obase = out + ((size_t)(bb * T_ + t) * L_) * DK_ + d;
#pragma unroll
    for (int v = 0; v < 8; ++v) {
        const int m = v + (lane >> 4) * 8;   // C layout: lanes16-31 hold M = v+8
        obase[(size_t)(l0 + m) * DK_] = acc[v] + bv;
    }
}

// ---------------------------------------------------------------------------
// Kernel 2: f[b,t,l] = x[b,l,:] . Wroot[t,:] + broot[t].  One block per (b,l);
// x row staged in LDS, 8 threads compute the 8 head dots.
// ---------------------------------------------------------------------------
__global__ __launch_bounds__(64)
void f_kernel(const float* __restrict__ x, const float* __restrict__ Wroot,
              const float* __restrict__ broot, float* __restrict__ f)
{
    __shared__ float xs[H_];
    const int bl = blockIdx.x;               // 0..B*L-1
    const int b  = bl / L_;
    const int l  = bl % L_;
    const float* xr = x + (size_t)bl * H_;
    for (int h = threadIdx.x; h < H_; h += 64) xs[h] = xr[h];
    __syncthreads();
    if (threadIdx.x < T_) {
        const int t = threadIdx.x;
        const float* w = Wroot + (size_t)t * H_;
        float s = 0.f;
        for (int h = 0; h < H_; ++h) s = fmaf(xs[h], w[h], s);
        f[(size_t)(b * T_ + t) * L_ + l] = s + broot[t];
    }
}

// ---------------------------------------------------------------------------
// Kernel 3: scores + mask + softmax + A = exp(p)*(1-I).
// One 256-thread block per (b,t,16-row stripe).  8 waves; each wave does four
// 16x16 WMMA column-tiles (K=96 in 24 steps of V_WMMA_F32_16X16X4_F32),
// stripe staged in LDS, row softmax via wave32 shuffles.
// ---------------------------------------------------------------------------
__global__ __launch_bounds__(256)
void scores_kernel(const float* __restrict__ q, const float* __restrict__ kmat,
                   const int* __restrict__ src_mask, float* __restrict__ A)
{
    __shared__ float S[16][L_];              // 32 KB
    const int si = blockIdx.x;               // row stripe
    const int t  = blockIdx.y;
    const int b  = blockIdx.z;
    const int bt = b * T_ + t;
    const int i0 = si * 16;
    const int tid  = threadIdx.x;
    const int wave = tid >> 5;
    const int lane = tid & 31;
    const int lm   = lane & 15;
    const int kh   = (lane >> 4) * 2;

    const float* qrow  = q    + (size_t)(bt * L_ + i0 + lm) * DK_;
    const float* kbase = kmat + (size_t)bt * L_ * DK_;
    const float  scale = rsqrtf((float)DK_);

    v8f acc[4];
#pragma unroll
    for (int c = 0; c < 4; ++c) acc[c] = (v8f){};

    for (int k0 = 0; k0 < DK_; k0 += 4) {
        v2f a;
        a.x = qrow[k0 + kh];
        a.y = qrow[k0 + kh + 1];
#pragma unroll
        for (int c = 0; c < 4; ++c) {
            const int j0 = (wave + 8 * c) * 16;
            const float* kr = kbase + (size_t)(j0 + lm) * DK_;
            v2f bb;
            bb.x = kr[k0 + kh];
            bb.y = kr[k0 + kh + 1];
            acc[c] = __builtin_amdgcn_wmma_f32_16x16x4_f32(false, a, false, bb,
                                                           (short)0, acc[c], false, false);
        }
    }
#pragma unroll
    for (int c = 0; c < 4; ++c) {
        const int j0 = (wave + 8 * c) * 16;
        const int j  = j0 + lm;
        const int mz = src_mask[b * L_ + j];
#pragma unroll
        for (int v = 0; v < 8; ++v) {
            const int row = v + (lane >> 4) * 8;
            float s = acc[c][v] * scale;
            if (mz == 0) s = -1.0e9f;
            S[row][j] = s;
        }
    }
    __syncthreads();

    // softmax per row; wave w handles rows 2w, 2w+1 of the stripe
    for (int rr = 0; rr < 2; ++rr) {
        const int r = wave * 2 + rr;
        const int i = i0 + r;
        float mx = -3.0e38f;
        for (int j = lane; j < L_; j += 32) mx = fmaxf(mx, S[r][j]);
        for (int o = 16; o > 0; o >>= 1) mx = fmaxf(mx, __shfl_xor(mx, o, 32));
        float sum = 0.f;
        for (int j = lane; j < L_; j += 32) sum += expf(S[r][j] - mx);
        for (int o = 16; o > 0; o >>= 1) sum += __shfl_xor(sum, o, 32);
        const float inv = 1.0f / sum;
        float* Arow = A + ((size_t)bt * L_ + i) * L_;
        for (int j = lane; j < L_; j += 32) {
            const float p = expf(S[r][j] - mx) * inv;
            Arow[j] = (j == i) ? 0.0f : expf(p);
        }
    }
}

// ---------------------------------------------------------------------------
// Kernel 4: deg[bt,j] = sum_i A[bt,i,j]  (column sums, coalesced over j)
// ---------------------------------------------------------------------------
__global__ __launch_bounds__(256)
void deg_kernel(const float* __restrict__ A, float* __restrict__ deg)
{
    const int idx = blockIdx.x * 256 + threadIdx.x;  // (bt, j)
    const int bt = idx / L_;
    const int j  = idx % L_;
    const float* Ab = A + (size_t)bt * L_ * L_;
    float s = 0.f;
    for (int i = 0; i < L_; ++i) {
        if (i + 8 < L_) __builtin_prefetch(&Ab[(size_t)(i + 8) * L_ + j], 0, 0);
        s += Ab[(size_t)i * L_ + j];
    }
    deg[idx] = s;
}

// ---------------------------------------------------------------------------
// Kernel 5: Lbar into d_out.  Row 0 = f; else -A + deg on diagonal.
// ---------------------------------------------------------------------------
__global__ __launch_bounds__(256)
void lbar_kernel(const float* __restrict__ A, const float* __restrict__ deg,
                 const float* __restrict__ f, float* __restrict__ M)
{
    const size_t idx = (size_t)blockIdx.x * 256 + threadIdx.x;
    const size_t bt  = idx / ((size_t)L_ * L_);
    const int rem = (int)(idx % ((size_t)L_ * L_));
    const int i = rem / L_;
    const int j = rem % L_;
    float v;
    if (i == 0) {
        v = f[bt * L_ + j];
    } else {
        v = -A[idx];
        if (i == j) v += deg[bt * L_ + j];
    }
    M[idx] = v;
}

// ---------------------------------------------------------------------------
// Kernel 6: blocked in-place Gauss-Jordan inversion, one 512x512 matrix per
// 1024-thread block.  Per 16-column block K = [k0,k0+16):
//   1) async-stage panel M[:,K] (512x16, stride-20 padded) and R = M[K,:]
//      into LDS via GLOBAL_LOAD_ASYNC_TO_LDS_B128 (+ s_wait_asynccnt)
//   2) factor panel in LDS with classic in-place GJ rules -> V = T[:,K]
//   3) full-width rank-16 update  M += (V - I[:,K]) @ R  via WMMA f32 16x16x4
//      (the K tile-column stores V directly).
// ---------------------------------------------------------------------------
__global__ __launch_bounds__(1024)
void invert_kernel(float* __restrict__ Mg)
{
    __shared__ __align__(16) float Pn[L_ * PSTR];   // 512 x 16 panel, stride 20
    __shared__ __align__(16) float R[NB_][L_];      // 16 x 512 pivot rows (original)
    float* M = Mg + (size_t)blockIdx.x * L_ * L_;
    const int tid  = threadIdx.x;
    const int wave = tid >> 5;               // 0..31
    const int lane = tid & 31;
    const int lm   = lane & 15;
    const int kh   = (lane >> 4) * 2;
    const int cpan = tid & 15;               // panel column owned in factorization
    const int ibas = tid >> 4;               // 0..63 base row

    for (int k0 = 0; k0 < L_; k0 += NB_) {
        // ---- async-stage panel and pivot-row block into LDS ----
        // panel: 512 rows x 16 floats, copied as 4-float (16B) chunks
        for (int e = tid; e < 2048; e += 1024) {
            const int i  = e >> 2;
            const int cc = (e & 3) * 4;
            const uint32_t lds = (uint32_t)(uintptr_t)&Pn[i * PSTR + cc];
            const float* g = M + (size_t)i * L_ + k0 + cc;
            asm volatile("global_load_async_to_lds_b128 %0, %1, off"
                         :: "v"(lds), "v"(g) : "memory");
        }
        // R: 16 rows x 512 floats, 16B chunks
        for (int e = tid; e < 2048; e += 1024) {
            const int p  = e >> 7;
            const int jc = (e & 127) * 4;
            const uint32_t lds = (uint32_t)(uintptr_t)&R[p][jc];
            const float* g = M + (size_t)(k0 + p) * L_ + jc;
            asm volatile("global_load_async_to_lds_b128 %0, %1, off"
                         :: "v"(lds), "v"(g) : "memory");
        }
        asm volatile("s_wait_asynccnt 0x0" ::: "memory");
        __syncthreads();

        // ---- in-LDS panel factorization: 16 classic GJ steps ----
        for (int p = 0; p < NB_; ++p) {
            const int g = k0 + p;            // global pivot row
            const float ipiv = 1.0f / Pn[g * PSTR + p];
            const float pgs  = Pn[g * PSTR + cpan] * ipiv;  // scaled pivot-row entry
            float fv[8], vv[8];
#pragma unroll
            for (int r = 0; r < 8; ++r) {
                const int i = ibas + 64 * r;
                fv[r] = Pn[i * PSTR + p];
                vv[r] = Pn[i * PSTR + cpan];
            }
            __syncthreads();
#pragma unroll
            for (int r = 0; r < 8; ++r) {
                const int i = ibas + 64 * r;
                float nv;
                if (i == g) nv = (cpan == p) ? ipiv : pgs;
                else        nv = (cpan == p) ? (-fv[r] * ipiv)
                                             : fmaf(-fv[r], pgs, vv[r]);
                Pn[i * PSTR + cpan] = nv;
            }
            __syncthreads();
        }

        // ---- rank-16 full-width update, wave w owns rows [16w, 16w+16) ----
        const int i0 = wave * 16;
        v2f afr[4];                          // U fragments, reused for 32 tiles
#pragma unroll
        for (int s = 0; s < 4; ++s) {
            const int kp = 4 * s + kh;
            const int gi = i0 + lm;
            afr[s].x = Pn[gi * PSTR + kp]     - ((gi == k0 + kp    ) ? 1.0f : 0.0f);
            afr[s].y = Pn[gi * PSTR + kp + 1] - ((gi == k0 + kp + 1) ? 1.0f : 0.0f);
        }
        const int ckol = k0 >> 4;            // tile column covering K
        for (int ct = 0; ct < 32; ++ct) {
            const int j = ct * 16 + lm;
            if (ct == ckol) {                // columns K <- V
#pragma unroll
                for (int v = 0; v < 8; ++v) {
                    const int row = v + (lane >> 4) * 8;
                    M[(size_t)(i0 + row) * L_ + j] = Pn[(i0 + row) * PSTR + lm];
                }
                continue;
            }
            v8f acc;
#pragma unroll
            for (int v = 0; v < 8; ++v) {
                const int row = v + (lane >> 4) * 8;
                acc[v] = M[(size_t)(i0 + row) * L_ + j];
            }
#pragma unroll
            for (int s = 0; s < 4; ++s) {
                const int kp = 4 * s + kh;
                v2f bfr;
                bfr.x = R[kp][j];
                bfr.y = R[kp + 1][j];
                acc = __builtin_amdgcn_wmma_f32_16x16x4_f32(false, afr[s], false, bfr,
                                                            (short)0, acc, false, false);
            }
#pragma unroll
            for (int v = 0; v < 8; ++v) {
                const int row = v + (lane >> 4) * 8;
                M[(size_t)(i0 + row) * L_ + j] = acc[v];
            }
        }
        __threadfence_block();
        __syncthreads();                     // block's writes visible before next stage
    }
}

// ---------------------------------------------------------------------------
// Kernel 7: att[bt,j,m] = (j!=0)*A[m,j]*LLinv[j,j] - (m!=0)*A[m,j]*LLinv[j,m]
// In-place over LLinv row j: reads before barrier, writes after.
// One block per (bt, j); only row j of LLinv is touched by this block.
// ---------------------------------------------------------------------------
__global__ __launch_bounds__(512)
void final_kernel(const float* __restrict__ A, float* __restrict__ M)
{
    __shared__ float s_diag;
    const int j  = blockIdx.x;
    const int bt = blockIdx.y;
    const int m  = threadIdx.x;
    float* Mrow = M + ((size_t)bt * L_ + j) * L_;
    const float llinv_jm = Mrow[m];
    if (m == 0) s_diag = Mrow[j];
    __syncthreads();
    const float amj = A[((size_t)bt * L_ + m) * L_ + j];
    float r = 0.f;
    if (j != 0) r = amj * s_diag;
    if (m != 0) r -= amj * llinv_jm;
    Mrow[m] = r;
}

// ---------------------------------------------------------------------------
extern "C" void kernel_launch(void* const* d_in, const int* in_sizes, int n_in,
                              void* d_out, int out_size, void* d_ws, size_t ws_size,
                              hipStream_t stream)
{
    const float* x        = (const float*)d_in[0];
    const int*   src_mask = (const int*)  d_in[1];
    const float* Wq       = (const float*)d_in[2];
    const float* bq       = (const float*)d_in[3];
    const float* Wk       = (const float*)d_in[4];
    const float* bk       = (const float*)d_in[5];
    const float* Wroot    = (const float*)d_in[6];
    const float* broot    = (const float*)d_in[7];

    float* ws = (float*)d_ws;
    const size_t qk1 = (size_t)B_ * T_ * L_ * DK_;     // 12,582,912 floats
    float* q    = ws;
    float* kbuf = ws + qk1;
    float* f    = ws + 2 * qk1;
    float* deg  = f + (size_t)B_ * T_ * L_;
    float* A    = deg + (size_t)B_ * T_ * L_;
    float* M    = (float*)d_out;                       // Lbar -> LLinv -> att, in place

    proj_kernel<<<dim3(H_ / 16, (B_ * L_) / 16, 2), 32, 0, stream>>>(
        x, Wq, bq, Wk, bk, q, kbuf);
    f_kernel<<<dim3(B_ * L_), 64, 0, stream>>>(x, Wroot, broot, f);
    scores_kernel<<<dim3(L_ / 16, T_, B_), 256, 0, stream>>>(q, kbuf, src_mask, A);
    deg_kernel<<<dim3((B_ * T_ * L_) / 256), 256, 0, stream>>>(A, deg);
    lbar_kernel<<<dim3((unsigned)(((size_t)B_ * T_ * L_ * L_) / 256)), 256, 0, stream>>>(
        A, deg, f, M);
    invert_kernel<<<dim3(B_ * T_), 1024, 0, stream>>>(M);
    final_kernel<<<dim3(L_, B_ * T_), 512, 0, stream>>>(A, M);
}